// pointnet_sa_module_32298154066787
// MI455X (gfx1250) — compile-verified
//
#include <hip/hip_runtime.h>

#define NPTS    2048
#define NPOINT  512
#define NSAMPLE 32
#define BATCH   16
#define C_IN    64
#define RAD2    0.04f   // 0.2^2

typedef float v2f __attribute__((ext_vector_type(2)));
typedef float v8f __attribute__((ext_vector_type(8)));

// ---------------------------------------------------------------------------
// Kernel 1: furthest point sampling -> writes new_xyz (B,3,512) directly.
// One workgroup per batch; dist[] lives in LDS; block argmax with JAX
// first-index tie-break. Emits the *previous* far each iteration (far0 = 0),
// matching the reference scan.
// ---------------------------------------------------------------------------
__global__ void __launch_bounds__(256)
fps_newxyz_kernel(const float* __restrict__ xyz, float* __restrict__ newxyz) {
  __shared__ float dist[NPTS];
  __shared__ float smax[256];
  __shared__ int   sidx[256];
  __shared__ int   sFar;
  const int b = blockIdx.x, tid = threadIdx.x;
  const float* px = xyz + (size_t)b * 3 * NPTS;
  const float* py = px + NPTS;
  const float* pz = py + NPTS;
  for (int n = tid; n < NPTS; n += 256) dist[n] = 1e10f;
  if (tid == 0) sFar = 0;
  __syncthreads();
  for (int i = 0; i < NPOINT; ++i) {
    const int far = sFar;
    if (tid < 3)
      newxyz[((size_t)b * 3 + tid) * NPOINT + i] =
          xyz[((size_t)b * 3 + tid) * NPTS + far];
    const float cx = px[far], cy = py[far], cz = pz[far];
    float bv = -1.0f; int bi = 0;
    for (int n = tid; n < NPTS; n += 256) {
      const float dx = px[n] - cx, dy = py[n] - cy, dz = pz[n] - cz;
      float d = dist[n];
      const float d2 = dx * dx + dy * dy + dz * dz;
      d = fminf(d, d2);
      dist[n] = d;
      if (d > bv) { bv = d; bi = n; }   // ascending n -> keeps first max
    }
    smax[tid] = bv; sidx[tid] = bi;
    __syncthreads();
    for (int off = 128; off > 0; off >>= 1) {
      if (tid < off) {
        const float ov = smax[tid + off]; const int oi = sidx[tid + off];
        if (ov > smax[tid] || (ov == smax[tid] && oi < sidx[tid])) {
          smax[tid] = ov; sidx[tid] = oi;
        }
      }
      __syncthreads();
    }
    if (tid == 0) sFar = sidx[0];
    __syncthreads();
  }
}

// ---------------------------------------------------------------------------
// Kernel 2: ball query. One wave32 per query point. Scans points in index
// order 32 at a time; ballot + prefix-popcount compaction preserves order
// (== taking the 32 smallest in-radius indices). Pad = smallest in-radius
// index, or N-1 when empty.
// ---------------------------------------------------------------------------
__global__ void __launch_bounds__(256)
ballquery_kernel(const float* __restrict__ xyz, const float* __restrict__ newxyz,
                 int* __restrict__ idxbuf) {
  const int wid  = (blockIdx.x * blockDim.x + threadIdx.x) >> 5;
  const int lane = threadIdx.x & 31;
  if (wid >= BATCH * NPOINT) return;
  const int b = wid >> 9, s = wid & (NPOINT - 1);
  const float* px = xyz + (size_t)b * 3 * NPTS;
  const float* py = px + NPTS;
  const float* pz = py + NPTS;
  const float qx = newxyz[((size_t)b * 3 + 0) * NPOINT + s];
  const float qy = newxyz[((size_t)b * 3 + 1) * NPOINT + s];
  const float qz = newxyz[((size_t)b * 3 + 2) * NPOINT + s];
  int* out = idxbuf + (size_t)wid * NSAMPLE;
  int count = 0;
  int minIdx = NPTS - 1;     // default pad when no point in radius
  bool haveMin = false;
  for (int base = 0; base < NPTS && count < NSAMPLE; base += 32) {
    const int n = base + lane;
    const float dx = px[n] - qx, dy = py[n] - qy, dz = pz[n] - qz;
    const bool pred = (dx * dx + dy * dy + dz * dz) < RAD2;
    const unsigned mask = (unsigned)__ballot(pred);
    if (!haveMin && mask) { minIdx = base + __ffs(mask) - 1; haveMin = true; }
    if (pred) {
      const int pos = count + __popc(mask & ((1u << lane) - 1u));
      if (pos < NSAMPLE) out[pos] = n;
    }
    count += __popc(mask);
  }
  if (count < NSAMPLE) {
    for (int slot = count + lane; slot < NSAMPLE; slot += 32) out[slot] = minIdx;
  }
}

// ---------------------------------------------------------------------------
// Kernel 3: fused gather + 3-layer MLP (fp32 WMMA) + max over NSAMPLE.
// Weights LDS-resident (64x68 + 128x64 + 256x128 = 177KB). Each group:
//   h0(68x32) -> relu(W1 h0 + b1)(64x32) -> relu(W2 . + b2)(128x32)
//   -> W3 . (256x32) -> max over 32 samples -> +b3, relu -> out(256).
// GEMMs use V_WMMA_F32_16X16X4_F32 with 16x16 output tiles per wave.
// ---------------------------------------------------------------------------
__global__ void __launch_bounds__(256)
mlp_fused_kernel(const float* __restrict__ xyz, const float* __restrict__ feat,
                 const int* __restrict__ idxbuf, const float* __restrict__ newxyz,
                 const float* __restrict__ W1, const float* __restrict__ b1,
                 const float* __restrict__ W2, const float* __restrict__ b2,
                 const float* __restrict__ W3, const float* __restrict__ b3,
                 float* __restrict__ outFeat) {
  extern __shared__ float smem[];
  float* sW1 = smem;               // 64 x 68 (K padded 67->68, zero fill)
  float* sW2 = sW1 + 64 * 68;      // 128 x 64
  float* sW3 = sW2 + 128 * 64;     // 256 x 128
  float* sH0 = sW3 + 256 * 128;    // 68 x 32
  float* sH1 = sH0 + 68 * 32;      // 64 x 32
  float* sH2 = sH1 + 64 * 32;      // 128 x 32
  const int tid   = threadIdx.x;
  const int lane  = tid & 31;
  const int wave  = tid >> 5;
  const int lrow  = lane & 15;     // position within lane half
  const int khalf = lane >> 4;     // 0: K {0,1}, 1: K {2,3} (A frag halves)

  for (int e = tid; e < 64 * 68; e += 256) {
    const int o = e / 68, c = e - o * 68;
    sW1[e] = (c < 67) ? W1[o * 67 + c] : 0.0f;
  }
  for (int e = tid; e < 128 * 64; e += 256)  sW2[e] = W2[e];
  for (int e = tid; e < 256 * 128; e += 256) sW3[e] = W3[e];
  __syncthreads();

  for (int g = blockIdx.x; g < BATCH * NPOINT; g += gridDim.x) {
    const int b = g >> 9, s = g & (NPOINT - 1);
    const int* gi = idxbuf + (size_t)g * NSAMPLE;
    // gather h0: rows 0-2 centered xyz, 3-66 features, 67 zero pad
    for (int e = tid; e < 68 * 32; e += 256) {
      const int row = e >> 5, k = e & 31;
      float v;
      if (row < 3) {
        const int n = gi[k];
        v = xyz[((size_t)b * 3 + row) * NPTS + n] -
            newxyz[((size_t)b * 3 + row) * NPOINT + s];
      } else if (row < 67) {
        const int n = gi[k];
        v = feat[((size_t)b * C_IN + (row - 3)) * NPTS + n];
      } else {
        v = 0.0f;
      }
      sH0[e] = v;
    }
    __syncthreads();

    // ---- layer 1: (64x68)x(68x32), 8 tile-tasks == 8 waves ----
    {
      const int mt = wave >> 1, nt = wave & 1;
      v8f acc = {};
      for (int k0 = 0; k0 < 68; k0 += 4) {
        v2f a, bb;
        const int ar = (mt * 16 + lrow) * 68 + k0 + 2 * khalf;
        a.x = sW1[ar]; a.y = sW1[ar + 1];
        const int bc = nt * 16 + lrow;
        bb.x = sH0[(k0 + 2 * khalf) * 32 + bc];
        bb.y = sH0[(k0 + 2 * khalf + 1) * 32 + bc];
        acc = __builtin_amdgcn_wmma_f32_16x16x4_f32(false, a, false, bb,
                                                    (short)0, acc, false, false);
      }
#pragma unroll
      for (int r = 0; r < 8; ++r) {
        const int row = mt * 16 + r + 8 * khalf;
        const int col = nt * 16 + lrow;
        const float v = acc[r] + b1[row];
        sH1[row * 32 + col] = v > 0.0f ? v : 0.0f;
      }
    }
    __syncthreads();

    // ---- layer 2: (128x64)x(64x32), 16 tile-tasks, 2 per wave ----
    for (int t = wave; t < 16; t += 8) {
      const int mt = t >> 1, nt = t & 1;
      v8f acc = {};
      for (int k0 = 0; k0 < 64; k0 += 4) {
        v2f a, bb;
        const int ar = (mt * 16 + lrow) * 64 + k0 + 2 * khalf;
        a.x = sW2[ar]; a.y = sW2[ar + 1];
        const int bc = nt * 16 + lrow;
        bb.x = sH1[(k0 + 2 * khalf) * 32 + bc];
        bb.y = sH1[(k0 + 2 * khalf + 1) * 32 + bc];
        acc = __builtin_amdgcn_wmma_f32_16x16x4_f32(false, a, false, bb,
                                                    (short)0, acc, false, false);
      }
#pragma unroll
      for (int r = 0; r < 8; ++r) {
        const int row = mt * 16 + r + 8 * khalf;
        const int col = nt * 16 + lrow;
        const float v = acc[r] + b2[row];
        sH2[row * 32 + col] = v > 0.0f ? v : 0.0f;
      }
    }
    __syncthreads();

    // ---- layer 3: (256x128)x(128x32) + fused max over 32 samples ----
    for (int mt = wave; mt < 16; mt += 8) {
      v8f acc0 = {}, acc1 = {};
      for (int k0 = 0; k0 < 128; k0 += 4) {
        v2f a, b0v, b1v;
        const int ar = (mt * 16 + lrow) * 128 + k0 + 2 * khalf;
        a.x = sW3[ar]; a.y = sW3[ar + 1];
        const int krow = (k0 + 2 * khalf) * 32;
        b0v.x = sH2[krow + lrow];       b0v.y = sH2[krow + 32 + lrow];
        b1v.x = sH2[krow + 16 + lrow];  b1v.y = sH2[krow + 32 + 16 + lrow];
        acc0 = __builtin_amdgcn_wmma_f32_16x16x4_f32(false, a, false, b0v,
                                                     (short)0, acc0, false, false);
        acc1 = __builtin_amdgcn_wmma_f32_16x16x4_f32(false, a, false, b1v,
                                                     (short)0, acc1, false, false);
      }
#pragma unroll
      for (int r = 0; r < 8; ++r) {
        // max over the two 16-sample halves, then across lanes in each half
        float m = fmaxf(acc0[r], acc1[r]);
        m = fmaxf(m, __shfl_xor(m, 1, 32));
        m = fmaxf(m, __shfl_xor(m, 2, 32));
        m = fmaxf(m, __shfl_xor(m, 4, 32));
        m = fmaxf(m, __shfl_xor(m, 8, 32));
        if (lrow == 0) {
          const int row = mt * 16 + r + 8 * khalf;
          const float v = m + b3[row];
          outFeat[((size_t)b * 256 + row) * NPOINT + s] = v > 0.0f ? v : 0.0f;
        }
      }
    }
    __syncthreads();
  }
}

extern "C" void kernel_launch(void* const* d_in, const int* in_sizes, int n_in,
                              void* d_out, int out_size, void* d_ws, size_t ws_size,
                              hipStream_t stream) {
  const float* xyz  = (const float*)d_in[0];  // (16,3,2048)
  const float* feat = (const float*)d_in[1];  // (16,64,2048)
  const float* W1   = (const float*)d_in[2];  // (64,67)
  const float* b1   = (const float*)d_in[3];  // (64)
  const float* W2   = (const float*)d_in[4];  // (128,64)
  const float* b2   = (const float*)d_in[5];  // (128)
  const float* W3   = (const float*)d_in[6];  // (256,128)
  const float* b3   = (const float*)d_in[7];  // (256)

  float* newxyz  = (float*)d_out;                       // (16,3,512)
  float* outFeat = newxyz + (size_t)BATCH * 3 * NPOINT; // (16,256,512)
  int*   idxbuf  = (int*)d_ws;                          // 8192*32 ints (1 MB)

  fps_newxyz_kernel<<<BATCH, 256, 0, stream>>>(xyz, newxyz);

  const int nQueries = BATCH * NPOINT;                  // 8192, one wave each
  ballquery_kernel<<<(nQueries * 32) / 256, 256, 0, stream>>>(xyz, newxyz, idxbuf);

  const size_t lds = (size_t)(64 * 68 + 128 * 64 + 256 * 128 +
                              68 * 32 + 64 * 32 + 128 * 32) * sizeof(float);
  mlp_fused_kernel<<<512, 256, lds, stream>>>(xyz, feat, idxbuf, newxyz,
                                              W1, b1, W2, b2, W3, b3, outFeat);
}